// PositionalEncodingFrame_26869315404024
// MI455X (gfx1250) — compile-verified
//
#include <hip/hip_runtime.h>

// out[row, :] = x[row, :] + pe[time_fra[row], :]
// rows = B*S = 65536, D = 512 fp32 (2 KB per row).
// Memory-bound: ~285 MB of HBM traffic -> ~12 us floor at 23.3 TB/s.
//
// CDNA5 paths used:
//  - global_load_async_to_lds_b128 (ASYNCcnt DMA) for the gathered pe row
//  - s_wait_asynccnt split counter
//  - TH_LOAD_NT / TH_STORE_NT on the once-streamed x/out (protects L2-resident pe)
//  - scalar s_load (KMcnt) for the block-uniform gather index

typedef float v4f __attribute__((ext_vector_type(4)));
typedef int   v4i __attribute__((vector_size(4 * sizeof(int))));

#define D_MODEL 512
#define N_ROWS  (16 * 4096)
#define TPB     128   // 128 threads x float4 = one 512-float row per block

__global__ __launch_bounds__(TPB) void pe_frame_add_kernel(
    const float* __restrict__ x,
    const int*   __restrict__ time_fra,
    const float* __restrict__ pe,
    float*       __restrict__ out)
{
    __shared__ float lds_pe[D_MODEL];   // 2 KB staging for the gathered pe row

    const unsigned row = blockIdx.x;
    const unsigned tid = threadIdx.x;

    // Uniform per block -> scalar load (KMcnt path).
    const unsigned t = (unsigned)time_fra[row];

    const v4f* __restrict__ x_row = (const v4f*)(x   + (size_t)row * D_MODEL);
    v4f*       __restrict__ o_row = (v4f*)      (out + (size_t)row * D_MODEL);
    float* pe_elem = const_cast<float*>(pe) + (size_t)t * D_MODEL + (size_t)tid * 4;

#if defined(__gfx1250__) && __has_builtin(__builtin_amdgcn_global_load_async_to_lds_b128)
    // CDNA5 async DMA: global -> LDS directly, no VGPR round trip (ASYNCcnt).
    __builtin_amdgcn_global_load_async_to_lds_b128(
        (__attribute__((address_space(1))) v4i*)pe_elem,
        (__attribute__((address_space(3))) v4i*)&lds_pe[tid * 4],
        /*imm offset*/0, /*cpol*/0);

    // Issue the streamed x load BEFORE draining ASYNCcnt so the pe DMA and the
    // x read are in flight simultaneously (both HBM latencies overlap).
    const v4f xv = __builtin_nontemporal_load(x_row + tid);

  #if __has_builtin(__builtin_amdgcn_s_wait_asynccnt)
    __builtin_amdgcn_s_wait_asynccnt(0);
  #else
    asm volatile("s_wait_asynccnt 0" ::: "memory");
  #endif
    // Each lane reads back exactly the 16 bytes it DMA'd -> per-wave wait is
    // sufficient, no __syncthreads needed.
    const v4f pv = *((const v4f*)lds_pe + tid);
#else
    // Fallback: direct global load of the gathered row (L2-resident table).
    const v4f pv = *((const v4f*)pe_elem);
    const v4f xv = __builtin_nontemporal_load(x_row + tid);
#endif

    __builtin_nontemporal_store(xv + pv, o_row + tid);
}

extern "C" void kernel_launch(void* const* d_in, const int* in_sizes, int n_in,
                              void* d_out, int out_size, void* d_ws, size_t ws_size,
                              hipStream_t stream) {
    // setup_inputs() order: x, time_fra, frame_emb (scratch, unused), pe
    const float* x        = (const float*)d_in[0];
    const int*   time_fra = (const int*)  d_in[1];
    const float* pe       = (const float*)d_in[3];
    float*       out      = (float*)d_out;

    dim3 grid(N_ROWS);
    dim3 block(TPB);
    pe_frame_add_kernel<<<grid, block, 0, stream>>>(x, time_fra, pe, out);
}